// PseudoImageScatter_17815524343997
// MI455X (gfx1250) — compile-verified
//
#include <hip/hip_runtime.h>
#include <hip/hip_bf16.h>
#include <stdint.h>

// Problem constants (from reference)
#define BATCH 4
#define NPILL 12000
#define FD    64
#define HH    496
#define WW    496
#define HW    ((size_t)HH * WW)

// ---------------------------------------------------------------------------
// Pass 1a: fill pillar-index map with -1
// ---------------------------------------------------------------------------
__global__ __launch_bounds__(256) void fill_neg1_kernel(int4* __restrict__ p, int n4) {
    int i = blockIdx.x * blockDim.x + threadIdx.x;
    if (i < n4) p[i] = make_int4(-1, -1, -1, -1);
}

// ---------------------------------------------------------------------------
// Pass 1b: scatter pillar index into (B,H,W) map. Duplicates resolved
// deterministically with atomicMax (JAX .set with dup indices is unspecified).
// ---------------------------------------------------------------------------
__global__ __launch_bounds__(256) void scatter_idx_kernel(const int* __restrict__ coords,
                                                          int* __restrict__ idxmap) {
    int i = blockIdx.x * blockDim.x + threadIdx.x;   // i = b*NPILL + p
    if (i >= BATCH * NPILL) return;
    int b = i / NPILL;
    int p = i - b * NPILL;
    const int* c = coords + (size_t)i * 4;
    int y = c[1];
    int x = c[2];
    if (x >= 0 && x < WW && y >= 0 && y < HH) {
        atomicMax(idxmap + ((size_t)b * HH + y) * WW + x, p);
    }
}

// ---------------------------------------------------------------------------
// Pass 2: gather + transpose. One block per (b, y) row.
//   - index row staged to LDS via TDM (tensor_load_to_lds) when available
//   - feature rows loaded coalesced into padded LDS tile, emitted transposed
// ---------------------------------------------------------------------------
struct SharedBlk {
    int   idx[512];          // must be FIRST member: TDM lds_addr assumes offset 0
    float feat[8][32 * 66];  // per-wave staging tile, stride 66 kills bank conflicts
};

__global__ __launch_bounds__(256) void gather_kernel(const float* __restrict__ feat,
                                                     const int* __restrict__ idxmap,
                                                     float* __restrict__ out) {
    __shared__ SharedBlk sh;

    const int by   = blockIdx.x;       // 0 .. BATCH*HH-1
    const int b    = by / HH;
    const int y    = by - b * HH;
    const int tid  = threadIdx.x;
    const int wave = tid >> 5;
    const int lane = tid & 31;

    const int* rowIdx = idxmap + ((size_t)b * HH + y) * WW;

#if __has_builtin(__builtin_amdgcn_tensor_load_to_lds)
    // --- Tensor Data Mover: DMA the 496-int index row into LDS (wave 0 only;
    //     TDM ignores EXEC, so gate at wave granularity). Descriptor per ISA
    //     cdna5 §8.3/8.4: 2D tile 496x1 of 4-byte elements, type=2 (image).
    typedef unsigned int u32x4 __attribute__((ext_vector_type(4)));
    typedef int          i32x8 __attribute__((ext_vector_type(8)));
    typedef int          i32x4 __attribute__((ext_vector_type(4)));
    if (wave == 0) {
        unsigned long long ga = (unsigned long long)(uintptr_t)rowIdx;
        u32x4 g0;
        g0[0] = 1u;                                       // count=1 (valid user desc)
        g0[1] = 0u;                                       // lds_addr = offset of sh.idx
        g0[2] = (unsigned)(ga & 0xFFFFFFFFull);           // global_addr[31:0]
        g0[3] = (unsigned)((ga >> 32) & 0x1FFFFFFull)     // global_addr[56:32]
                | (2u << 30);                             // type = 2
        i32x8 g1;
        g1[0] = 0x00020000;                // data_size = 2 (4 bytes); mask/flags 0
        g1[1] = (WW & 0xFFFF) << 16;       // tensor_dim0 lo16 = 496
        g1[2] = (1 << 16);                 // tensor_dim0 hi=0 | tensor_dim1 lo16 = 1
        g1[3] = (WW & 0xFFFF) << 16;       // tensor_dim1 hi=0 | tile_dim0 = 496
        g1[4] = 1;                         // tile_dim1 = 1, tile_dim2 = 0
        g1[5] = WW;                        // tensor_dim0_stride lo32 = 496
        g1[6] = 0;
        g1[7] = 0;
        i32x4 z4 = {0, 0, 0, 0};
        i32x8 z8 = {0, 0, 0, 0, 0, 0, 0, 0};
        __builtin_amdgcn_tensor_load_to_lds(g0, g1, z4, z4, z8, 0);
#if __has_builtin(__builtin_amdgcn_s_wait_tensorcnt)
        __builtin_amdgcn_s_wait_tensorcnt(0);
#else
        asm volatile("s_wait_tensorcnt 0x0" ::: "memory");
#endif
    }
#else
    for (int i = tid; i < WW; i += 256) sh.idx[i] = rowIdx[i];
#endif
    __syncthreads();

    const float* featB = feat + (size_t)b * NPILL * FD;
    float*       outB  = out + (size_t)b * FD * HW + (size_t)y * WW;

    // Every wave executes exactly 2 chunk iterations (496 = 8 waves * 32 * 2 - 16),
    // so the in-loop __syncthreads() is uniform.
    for (int x0 = wave * 32; x0 < WW; x0 += 256) {
        const int nx = (WW - x0 < 32) ? (WW - x0) : 32;
        float* sf = sh.feat[wave];

        // Stage phase: coalesced 256B row loads (float2 per lane), zeros if empty
        for (int i = 0; i < nx; ++i) {
            int p = sh.idx[x0 + i];
            float2 v = make_float2(0.0f, 0.0f);
            if (p >= 0) {
                const float2* src = (const float2*)(featB + (size_t)p * FD);
                v = src[lane];                       // f = 2*lane, 2*lane+1
            }
            sf[i * 66 + 2 * lane]     = v.x;
            sf[i * 66 + 2 * lane + 1] = v.y;
        }
        __syncthreads();   // order LDS stores before cross-lane LDS reads

        // Emit phase: lane <-> x, loop over channels. 32 consecutive dwords per
        // wave per channel -> fully coalesced writes; LDS reads conflict-free
        // (addresses lane*66+f hit 32 distinct banks).
        if (lane < nx) {
            float* dst = outB + x0 + lane;
            for (int f = 0; f < FD; ++f) {
                dst[(size_t)f * HW] = sf[lane * 66 + f];
            }
        }
        __syncthreads();   // keep iteration structure uniform across waves
    }
}

// ---------------------------------------------------------------------------
// Fallback path (only if workspace cannot hold the 3.9 MB index map)
// ---------------------------------------------------------------------------
__global__ __launch_bounds__(256) void zero_out_kernel(float4* __restrict__ p, int n4) {
    int i = blockIdx.x * blockDim.x + threadIdx.x;
    if (i < n4) p[i] = make_float4(0.f, 0.f, 0.f, 0.f);
}

__global__ __launch_bounds__(256) void direct_scatter_kernel(const float* __restrict__ feat,
                                                             const int* __restrict__ coords,
                                                             float* __restrict__ out) {
    int i = blockIdx.x * blockDim.x + threadIdx.x;   // i = b*NPILL + p
    if (i >= BATCH * NPILL) return;
    int b = i / NPILL;
    const int* c = coords + (size_t)i * 4;
    int y = c[1];
    int x = c[2];
    if (x >= 0 && x < WW && y >= 0 && y < HH) {
        const float* src = feat + (size_t)i * FD;
        float* dst = out + (size_t)b * FD * HW + (size_t)y * WW + x;
        for (int f = 0; f < FD; ++f) dst[(size_t)f * HW] = src[f];
    }
}

// ---------------------------------------------------------------------------
extern "C" void kernel_launch(void* const* d_in, const int* in_sizes, int n_in,
                              void* d_out, int out_size, void* d_ws, size_t ws_size,
                              hipStream_t stream) {
    const float* feat   = (const float*)d_in[0];
    const int*   coords = (const int*)d_in[1];
    float*       out    = (float*)d_out;

    const size_t mapInts  = (size_t)BATCH * HH * WW;            // 984,064
    const size_t mapBytes = mapInts * sizeof(int);              // ~3.9 MB

    if (ws_size >= mapBytes) {
        int* idxmap = (int*)d_ws;

        int n4 = (int)(mapInts / 4);
        fill_neg1_kernel<<<(n4 + 255) / 256, 256, 0, stream>>>((int4*)idxmap, n4);

        int np = BATCH * NPILL;
        scatter_idx_kernel<<<(np + 255) / 256, 256, 0, stream>>>(coords, idxmap);

        gather_kernel<<<BATCH * HH, 256, 0, stream>>>(feat, idxmap, out);
    } else {
        int n4 = out_size / 4;
        zero_out_kernel<<<(n4 + 255) / 256, 256, 0, stream>>>((float4*)out, n4);

        int np = BATCH * NPILL;
        direct_scatter_kernel<<<(np + 255) / 256, 256, 0, stream>>>(feat, coords, out);
    }
}